// RBF_67430986547312
// MI455X (gfx1250) — compile-verified
//
#include <hip/hip_runtime.h>
#include <hip/hip_bf16.h>
#include <cstddef>

typedef float v2f __attribute__((ext_vector_type(2)));
typedef float v8f __attribute__((ext_vector_type(8)));

#define RBF_D 32   // feature dim; total GEMM K = 64 (x^2 block + x block)

// ---------------------------------------------------------------------------
// Prep: build fragment-ready B-pairs and fused scale.
//   Bp[p][m] = (B'[2p][m], B'[2p+1][m]),  p = 0..31  (B' is 64 x M)
//     rows 0..31  of B' : 1/sigma^2       (coeff of x^2)
//     rows 32..63 of B' : -2*c/sigma^2    (coeff of x)
//   scale[m] = exp(-0.5*(sum c^2/s2 + D*log(2pi) + sum log s2))
//            = exp(-0.5*bias) / detS
// ---------------------------------------------------------------------------
__global__ void rbf_prep(const float* __restrict__ centers,
                         const float* __restrict__ sigma,
                         v2f*  __restrict__ Bp,      // [D][M] float2 pairs
                         float* __restrict__ scale,  // [M]
                         int M) {
    int m = blockIdx.x * blockDim.x + threadIdx.x;
    if (m >= M) return;
    float bias = 0.0f;
    float ls   = 0.0f;
    #pragma unroll
    for (int d = 0; d < RBF_D; d += 2) {
        float c0 = centers[(size_t)m * RBF_D + d];
        float c1 = centers[(size_t)m * RBF_D + d + 1];
        float s0 = sigma[(size_t)m * RBF_D + d];
        float s1 = sigma[(size_t)m * RBF_D + d + 1];
        float s20 = s0 * s0, s21 = s1 * s1;
        float i0 = 1.0f / s20, i1 = 1.0f / s21;
        v2f w1; w1.x = i0;              w1.y = i1;
        v2f w2; w2.x = -2.0f * c0 * i0; w2.y = -2.0f * c1 * i1;
        Bp[(size_t)(d / 2) * M + m]              = w1;  // pair p = d/2
        Bp[(size_t)(RBF_D / 2 + d / 2) * M + m]  = w2;  // pair p = 16 + d/2
        bias += c0 * c0 * i0 + c1 * c1 * i1;
        ls   += __logf(s20) + __logf(s21);
    }
    const float LOG2PI = 1.8378770664093453f;
    scale[m] = __expf(-0.5f * (bias + (float)RBF_D * LOG2PI + ls));
}

// ---------------------------------------------------------------------------
// Main GEMM + epilogue. Block = 128 threads = 4 waves.
// Wave w: rows [rowBase, rowBase+32) as two 16-row strips,
//         cols [colBase, colBase+64) as four 16-col tiles.
// 16 k-steps x (2 strips x 4 tiles) = 128 v_wmma_f32_16x16x4_f32 per wave.
// Each B fragment (one b64 load) feeds 2 WMMAs.
// ---------------------------------------------------------------------------
__global__ __launch_bounds__(128)
void rbf_wmma(const float* __restrict__ x,
              const v2f*  __restrict__ Bp,
              const float* __restrict__ scale,
              float* __restrict__ out,
              int N, int M) {
    const int lane = threadIdx.x & 31;
    const int w    = threadIdx.x >> 5;
    const int r    = lane & 15;   // row-in-tile / col-in-tile
    const int h    = lane >> 4;   // half select
    const size_t Ms = (size_t)M;

    const int rowBase = blockIdx.x * 128 + w * 32;
    const int colBase = blockIdx.y * 64;
    if (rowBase >= N || colBase >= M) return;  // wave-uniform

    // --- x fragments for both strips: lane (r,h) needs cols {4t+2h, 4t+2h+1}.
    float xa[2][16];
    #pragma unroll
    for (int s = 0; s < 2; ++s) {
        const float* xrow = x + (size_t)(rowBase + 16 * s + r) * RBF_D;
        #pragma unroll
        for (int t = 0; t < 8; ++t) {
            v2f v = *reinterpret_cast<const v2f*>(xrow + 4 * t + 2 * h);
            xa[s][2 * t]     = v.x;
            xa[s][2 * t + 1] = v.y;
        }
    }

    // --- Zero-init accumulators (bias folded into scale[m]).
    v8f acc[2][4];
    const v8f vzero = {0.f, 0.f, 0.f, 0.f, 0.f, 0.f, 0.f, 0.f};
    #pragma unroll
    for (int s = 0; s < 2; ++s)
        #pragma unroll
        for (int tile = 0; tile < 4; ++tile) acc[s][tile] = vzero;

    // Fragment pointer: pair index p = 2t + h, element offset p*M + (colBase+16*tile+r).
    // Tile offsets are immediate (128*tile bytes); t advances via one pointer add.
    const v2f* __restrict__ bptr = Bp + (size_t)h * Ms + (size_t)(colBase + r);

    // --- Pass 1: A = x^2 against pairs p = 0..15 (the 1/sigma^2 block).
    #pragma unroll
    for (int t = 0; t < 8; ++t) {
        v2f bf[4];
        #pragma unroll
        for (int tile = 0; tile < 4; ++tile) bf[tile] = bptr[16 * tile];
        v2f a0, a1;
        a0.x = xa[0][2 * t] * xa[0][2 * t];  a0.y = xa[0][2 * t + 1] * xa[0][2 * t + 1];
        a1.x = xa[1][2 * t] * xa[1][2 * t];  a1.y = xa[1][2 * t + 1] * xa[1][2 * t + 1];
        #pragma unroll
        for (int tile = 0; tile < 4; ++tile) {
            acc[0][tile] = __builtin_amdgcn_wmma_f32_16x16x4_f32(
                false, a0, false, bf[tile], (short)0, acc[0][tile], false, false);
            acc[1][tile] = __builtin_amdgcn_wmma_f32_16x16x4_f32(
                false, a1, false, bf[tile], (short)0, acc[1][tile], false, false);
        }
        bptr += 2 * Ms;
    }

    // --- Pass 2: A = x against pairs p = 16..31 (the -2c/sigma^2 block).
    #pragma unroll
    for (int t = 0; t < 8; ++t) {
        v2f bf[4];
        #pragma unroll
        for (int tile = 0; tile < 4; ++tile) bf[tile] = bptr[16 * tile];
        v2f a0, a1;
        a0.x = xa[0][2 * t];  a0.y = xa[0][2 * t + 1];
        a1.x = xa[1][2 * t];  a1.y = xa[1][2 * t + 1];
        #pragma unroll
        for (int tile = 0; tile < 4; ++tile) {
            acc[0][tile] = __builtin_amdgcn_wmma_f32_16x16x4_f32(
                false, a0, false, bf[tile], (short)0, acc[0][tile], false, false);
            acc[1][tile] = __builtin_amdgcn_wmma_f32_16x16x4_f32(
                false, a1, false, bf[tile], (short)0, acc[1][tile], false, false);
        }
        bptr += 2 * Ms;
    }

    // --- Epilogue: out = exp(-0.5*acc)*scale = exp2(acc * (-0.5*log2 e)) * scale.
    // __builtin_amdgcn_exp2f = raw v_exp_f32 (same native path __expf uses),
    // with the -0.5 and log2(e) factors pre-folded into one constant:
    // one v_mul + one v_exp_f32 + one v_mul per output.
    const float NHALF_LOG2E = -0.72134752044448170367f;  // -0.5 * log2(e)
    float sc[4];
    #pragma unroll
    for (int tile = 0; tile < 4; ++tile) sc[tile] = scale[colBase + 16 * tile + r];

    #pragma unroll
    for (int s = 0; s < 2; ++s) {
        float* optr = out + (size_t)(rowBase + 16 * s + 8 * h) * Ms + (colBase + r);
        #pragma unroll
        for (int v = 0; v < 8; ++v) {
            #pragma unroll
            for (int tile = 0; tile < 4; ++tile) {
                __builtin_nontemporal_store(
                    __builtin_amdgcn_exp2f(acc[s][tile][v] * NHALF_LOG2E) * sc[tile],
                    optr + 16 * tile);
            }
            optr += Ms;
        }
    }
}

// ---------------------------------------------------------------------------
extern "C" void kernel_launch(void* const* d_in, const int* in_sizes, int n_in,
                              void* d_out, int out_size, void* d_ws, size_t ws_size,
                              hipStream_t stream) {
    const float* x       = (const float*)d_in[0];
    const float* centers = (const float*)d_in[1];
    const float* sigma   = (const float*)d_in[2];
    float*       out     = (float*)d_out;

    const int N = in_sizes[0] / RBF_D;   // 16384
    const int M = in_sizes[1] / RBF_D;   // 1024

    // Workspace: Bp [D x M] float2 (256 KB) | scale [M] (4 KB)
    v2f*   Bp    = (v2f*)d_ws;
    float* scale = (float*)(Bp + (size_t)RBF_D * M);

    rbf_prep<<<(M + 127) / 128, 128, 0, stream>>>(centers, sigma, Bp, scale, M);

    dim3 grid((N + 127) / 128, (M + 63) / 64);
    rbf_wmma<<<grid, 128, 0, stream>>>(x, Bp, scale, out, N, M);
}